// DCConv3dKernelPolynomials_12317966205075
// MI455X (gfx1250) — compile-verified
//
#include <hip/hip_runtime.h>
#include <hip/hip_bf16.h>
#include <math.h>

typedef __attribute__((ext_vector_type(2))) float v2f;
typedef __attribute__((ext_vector_type(8))) float v8f;

#define NPTS   32768   // 8 * 4096 points  (N dimension)
#define MROWS  4096    // 64 * 64 channels (M dimension)
#define NK     23
#define NKPAD  24
#define NT2    (NPTS / 32)    // 1024 N super-tiles (32 wide)
#define MT2    (MROWS / 32)   // 128  M super-tiles (32 tall)

// ---------------------------------------------------------------------------
// Stage 0: pad coefficients (4096 x 23) -> (4096 x 24), zero-filled tail.
// ---------------------------------------------------------------------------
__global__ __launch_bounds__(256) void prep_coeff(const float* __restrict__ coeff,
                                                  float* __restrict__ coeffPad) {
    int t = blockIdx.x * blockDim.x + threadIdx.x;
    if (t >= MROWS * NKPAD) return;
    int mn = t / NKPAD;
    int k  = t - mn * NKPAD;
    coeffPad[t] = (k < NK) ? coeff[mn * NK + k] : 0.0f;
}

// ---------------------------------------------------------------------------
// Stage 1: per-point hydrogen wavefunction basis, (32768 x 24) k-major.
// Combo table is constexpr so full unrolling constant-folds every factorial,
// Laguerre and Legendre coefficient at compile time.
// ---------------------------------------------------------------------------
__device__ __forceinline__ float factf(int k) {
    float f = 1.0f;
    #pragma unroll
    for (int i = 2; i <= 8; ++i) if (i <= k) f *= (float)i;
    return f;
}
__device__ __forceinline__ float combf(int p, int q) {
    return factf(p) / (factf(q) * factf(p - q));
}

__device__ __forceinline__ float laguerre(int k, int a, float x) {
    // L_k^a(x), k <= 3
    float out = 0.0f, xp = 1.0f;
    #pragma unroll
    for (int i = 0; i <= 3; ++i) {
        if (i <= k) {
            float c = ((i & 1) ? -1.0f : 1.0f) * combf(k + a, k - i) / factf(i);
            out = fmaf(c, xp, out);
        }
        xp *= x;
    }
    return out;
}

__device__ __forceinline__ float plegendre(int l, int m, float ct, float st) {
    // P_l^m, l <= 2, 0 <= m <= l, Condon-Shortley phase
    float pmm = 1.0f;
    if (m > 0) {
        float df = 1.0f;
        #pragma unroll
        for (int i = 1; i < 5; i += 2) if (i < 2 * m) df *= (float)i;
        float stm = 1.0f;
        #pragma unroll
        for (int i = 0; i < 2; ++i) if (i < m) stm *= st;
        pmm = ((m & 1) ? -df : df) * stm;
    }
    if (l == m) return pmm;
    float pmmp1 = ct * (float)(2 * m + 1) * pmm;
    if (l == m + 1) return pmmp1;
    float pll = pmmp1;
    #pragma unroll
    for (int ll = 2; ll <= 2; ++ll) {
        if (ll >= m + 2 && ll <= l) {
            pll = ((float)(2 * ll - 1) * ct * pmmp1 - (float)(ll + m - 1) * pmm) / (float)(ll - m);
            pmm = pmmp1; pmmp1 = pll;
        }
    }
    return pll;
}

__global__ __launch_bounds__(256) void prep_poly(const float* __restrict__ pos,
                                                 float* __restrict__ polyPad) {
    int p = blockIdx.x * blockDim.x + threadIdx.x;
    if (p >= NPTS) return;

    float x = pos[3 * p + 0];
    float y = pos[3 * p + 1];
    float z = pos[3 * p + 2];
    float r  = sqrtf(x * x + y * y + z * z) + 1e-12f;
    float ct = z / r;
    float st2 = 1.0f - ct * ct;
    float st  = sqrtf(st2 > 0.0f ? st2 : 0.0f);
    float phi = atan2f(y, x);

    constexpr int combos[23][3] = {
        {1,0,0},
        {2,0,0},{2,1,-1},{2,1,0},{2,1,1},
        {3,0,0},{3,1,-1},{3,1,0},{3,1,1},{3,2,-2},{3,2,-1},{3,2,0},{3,2,1},{3,2,2},
        {4,0,0},{4,1,-1},{4,1,0},{4,1,1},{4,2,-2},{4,2,-1},{4,2,0},{4,2,1},{4,2,2}
    };
    constexpr float PI4 = 4.0f * 3.14159265358979323846f;
    constexpr float SQRT2 = 1.41421356237309504880f;

    float* dst = polyPad + (size_t)p * NKPAD;

    #pragma unroll
    for (int c = 0; c < NK; ++c) {
        const int n = combos[c][0], l = combos[c][1], m = combos[c][2];
        const float fn = (float)n;
        float rho = 2.0f * r / fn;
        float norm_r = sqrtf((8.0f / (fn * fn * fn)) * factf(n - l - 1)
                             / (2.0f * fn * factf(n + l)));
        float rl = 1.0f;
        #pragma unroll
        for (int i = 0; i < 2; ++i) if (i < l) rl *= rho;
        float R = norm_r * expf(-0.5f * rho) * rl * laguerre(n - l - 1, 2 * l + 1, rho);
        const int am = (m < 0) ? -m : m;
        float Klm = sqrtf((float)(2 * l + 1) / PI4 * factf(l - am) / factf(l + am));
        float P = plegendre(l, am, ct, st);
        float Y;
        if (m > 0)      Y = SQRT2 * Klm * cosf((float)am * phi) * P;
        else if (m < 0) Y = SQRT2 * Klm * sinf((float)am * phi) * P;
        else            Y = Klm * P;
        dst[c] = R * Y;
    }
    dst[23] = 0.0f;  // K padding
}

// ---------------------------------------------------------------------------
// Stage 2: out[mn][p] = sum_k coeffPad[mn][k] * polyPad[p][k]
// One wave per 32x32 super-tile: 2x2 register blocking of 16x16 WMMA tiles,
// 24 x V_WMMA_F32_16X16X4_F32 per wave (K = 24 in 6 steps of 4).
// Operand traffic: 6 KB loaded per 4 KB stored (vs 3 KB per 1 KB unblocked),
// keeping L2 read traffic (~768 MB total) under the 512 MB HBM store floor.
//
// Fragment layout (32-bit, 16x4 A / 4x16 B per ISA 7.12.2):
//   lanes 0-15 : K = k0+0 (VGPR0), k0+1 (VGPR1)
//   lanes 16-31: K = k0+2 (VGPR0), k0+3 (VGPR1)
// -> each lane loads one aligned float2 of consecutive-k values.
// C/D layout: VGPR v -> row (v + 8*hi), lane&15 -> col.
// ---------------------------------------------------------------------------
__device__ __forceinline__ v2f load_frag(const float* __restrict__ base) {
    const float2 f = *(const float2*)base;
    v2f v; v.x = f.x; v.y = f.y;
    return v;
}

__global__ __launch_bounds__(256) void gemm_wmma(const float* __restrict__ coeffPad,
                                                 const float* __restrict__ polyPad,
                                                 float* __restrict__ out) {
    const int wave = blockIdx.x * 8 + (threadIdx.x >> 5);
    const int lane = threadIdx.x & 31;
    const int nt = wave & (NT2 - 1);   // fast index: waves in a block share A tiles
    const int mt = wave >> 10;         // NT2 == 1024 == 1<<10
    const int Mbase = mt * 32;
    const int Nbase = nt * 32;
    const int lo = lane & 15;
    const int hi = lane >> 4;

    const float* aRow0 = coeffPad + (size_t)(Mbase + lo) * NKPAD;       // M tile 0
    const float* aRow1 = aRow0 + 16 * NKPAD;                            // M tile 1
    const float* bRow0 = polyPad  + (size_t)(Nbase + lo) * NKPAD;       // N tile 0
    const float* bRow1 = bRow0 + 16 * NKPAD;                            // N tile 1

    v8f c00 = {}, c01 = {}, c10 = {}, c11 = {};
    #pragma unroll
    for (int s = 0; s < 6; ++s) {
        const int ka = s * 4 + hi * 2;
        const v2f a0 = load_frag(aRow0 + ka);
        const v2f a1 = load_frag(aRow1 + ka);
        const v2f b0 = load_frag(bRow0 + ka);
        const v2f b1 = load_frag(bRow1 + ka);
        c00 = __builtin_amdgcn_wmma_f32_16x16x4_f32(false, a0, false, b0, (short)0, c00, false, false);
        c01 = __builtin_amdgcn_wmma_f32_16x16x4_f32(false, a0, false, b1, (short)0, c01, false, false);
        c10 = __builtin_amdgcn_wmma_f32_16x16x4_f32(false, a1, false, b0, (short)0, c10, false, false);
        c11 = __builtin_amdgcn_wmma_f32_16x16x4_f32(false, a1, false, b1, (short)0, c11, false, false);
    }

    // Store 32x32 f32 tile; each instruction covers 2 x 64B contiguous segments.
    float* o00 = out + (size_t)(Mbase + hi * 8) * NPTS + Nbase + lo;
    #pragma unroll
    for (int v = 0; v < 8; ++v) {
        float* rowp = o00 + (size_t)v * NPTS;
        __builtin_nontemporal_store(c00[v], rowp);                         // rows M+0..15,  cols N+0..15
        __builtin_nontemporal_store(c01[v], rowp + 16);                    // rows M+0..15,  cols N+16..31
        __builtin_nontemporal_store(c10[v], rowp + (size_t)16 * NPTS);     // rows M+16..31, cols N+0..15
        __builtin_nontemporal_store(c11[v], rowp + (size_t)16 * NPTS + 16);// rows M+16..31, cols N+16..31
    }
}

// ---------------------------------------------------------------------------
extern "C" void kernel_launch(void* const* d_in, const int* in_sizes, int n_in,
                              void* d_out, int out_size, void* d_ws, size_t ws_size,
                              hipStream_t stream) {
    const float* pos   = (const float*)d_in[0];   // (8, 4096, 3) f32
    const float* coeff = (const float*)d_in[1];   // (64, 64, 23) f32
    float* out = (float*)d_out;                   // (64, 64, 8, 4096) f32

    float* coeffPad = (float*)d_ws;               //  4096 * 24 f32 = 384 KiB
    float* polyPad  = coeffPad + MROWS * NKPAD;   // 32768 * 24 f32 =   3 MiB

    prep_coeff<<<(MROWS * NKPAD + 255) / 256, 256, 0, stream>>>(coeff, coeffPad);
    prep_poly <<<(NPTS + 255) / 256, 256, 0, stream>>>(pos, polyPad);

    // 128 * 1024 super-tiles, 8 waves per 256-thread block
    gemm_wmma<<<(MT2 * NT2) / 8, 256, 0, stream>>>(coeffPad, polyPad, out);
}